// AVRRender_79585743995419
// MI455X (gfx1250) — compile-verified
//
#include <hip/hip_runtime.h>
#include <hip/hip_bf16.h>
#include <stdint.h>

// ---------------------------------------------------------------------------
// AVR render for MI455X (gfx1250, wave32).
//
//   k1 : per-(b,ray) weights w[b,r,s] (cumprod over 64 samples) + delay index
//        (ray directions via exact Threefry2x32(key=42) replication).
//   k2 : memory-bound weighted ray reduction of signal (302 MB stream,
//        the whole-problem roofline term: ~13 us @ 23.3 TB/s).
//   k3 : reduce 16 partial chunks, apply tail mask + path loss -> H[128][512].
//   k3b: precompute DFT trig matrix TRIG[512][528] once (shared by all tiles).
//   k4 : DFT as a real GEMM on the matrix core (V_WMMA_F32_16X16X4_F32):
//        F[128 x 514] = H[128 x 512] x TRIG[512 x 514]  (cos cols | sin cols)
//        Inner loop = loads + WMMA only (trig hoisted out).
//   k5 : phase rotation + sum over samples -> receive (2 x 257 x {re,im}).
// ---------------------------------------------------------------------------

#define BS      2
#define NR      1154
#define NS      64
#define LSEQ    512
#define LH      257          // LSEQ/2 + 1
#define NCOL    (2*LH)       // 514: [0,257) cos sums, [257,514) sin sums
#define NCOLP   528          // padded to 33*16 for WMMA tiles
#define RCHUNK  16
#define RPC     73           // ceil(NR / RCHUNK)

#define FS_C    16000.0f
#define SPD_C   343.0f
#define TWOPI   6.28318530717958647692f

typedef __attribute__((ext_vector_type(2))) float v2f;
typedef __attribute__((ext_vector_type(8))) float v8f;

// workspace byte offsets
#define OFF_W    ((size_t)0)                         // BS*NR*NS f32 = 590848 B
#define OFF_DLY  ((size_t)590848)                    // BS*NR*NS i32 = 590848 B
#define OFF_PART ((size_t)1181696)                   // BS*NS*RCHUNK*LSEQ f32 = 4 MB
#define OFF_H    ((size_t)5376000)                   // 128*512 f32 = 262144 B
#define OFF_F    ((size_t)5638144)                   // 128*514 f32 = 263168 B
#define OFF_TRIG ((size_t)5901312)                   // 512*528 f32 = 1081344 B
// total ~6.98 MB

__device__ __forceinline__ float dval(int s) {
    return 0.2f + (9.8f / 63.0f) * (float)s;         // linspace(0,1,64)*(FAR-NEAR)+NEAR
}

// --- Threefry-2x32, matching jax.random (5 groups of 4 rounds) -------------
__device__ __forceinline__ uint32_t rotl32(uint32_t x, int d) {
    return (x << d) | (x >> (32 - d));
}
__device__ __forceinline__ void threefry2x32(uint32_t k0, uint32_t k1,
                                             uint32_t c0, uint32_t c1,
                                             uint32_t* o0, uint32_t* o1) {
    uint32_t ks[3] = { k0, k1, k0 ^ k1 ^ 0x1BD11BDAu };
    uint32_t x0 = c0 + ks[0];
    uint32_t x1 = c1 + ks[1];
    const int rotA[4] = {13, 15, 26, 6};
    const int rotB[4] = {17, 29, 16, 24};
    #pragma unroll
    for (int i = 0; i < 5; ++i) {
        const int* R = (i & 1) ? rotB : rotA;
        #pragma unroll
        for (int j = 0; j < 4; ++j) { x0 += x1; x1 = rotl32(x1, R[j]); x1 ^= x0; }
        x0 += ks[(i + 1) % 3];
        x1 += ks[(i + 2) % 3] + (uint32_t)(i + 1);
    }
    *o0 = x0; *o1 = x1;
}

// === k1: weights + delays (one thread per (b, ray)) ========================
__global__ void k1_weights(const float* __restrict__ rays_o,
                           const float* __restrict__ tx,
                           const float* __restrict__ attn,
                           float* __restrict__ W, int* __restrict__ DLY) {
    int idx = blockIdx.x * blockDim.x + threadIdx.x;
    if (idx >= BS * NR) return;
    int b = idx / NR, r = idx % NR;

    // ray direction (jax.random.uniform(key(42),(48,)) azimuth jitter)
    float dx, dy, dz;
    if (r >= NR - 2) {
        dx = 0.f; dy = 0.f; dz = (r == NR - 2) ? 1.f : -1.f;
    } else {
        int ia = r / 24, ie = r % 24;              // meshgrid 'ij': azi-major
        uint32_t c0, c1; bool lo = (ia < 24);
        if (lo) { c0 = (uint32_t)ia;        c1 = (uint32_t)(ia + 24); }
        else    { c0 = (uint32_t)(ia - 24); c1 = (uint32_t)ia; }
        uint32_t o0, o1;
        threefry2x32(0u, 42u, c0, c1, &o0, &o1);   // jax.random.key(42) = {0,42}
        uint32_t bits = lo ? o0 : o1;
        float u = __uint_as_float((bits >> 9) | 0x3f800000u) - 1.0f;  // [0,1)
        float azi = ((float)ia + u) * (TWOPI / 48.0f);
        float ele = acosf(2.0f * ((float)(ie + 1) / 25.0f) - 1.0f);
        float se = sinf(ele);
        dx = cosf(azi) * se; dy = sinf(azi) * se; dz = cosf(ele);
    }

    float ox = rays_o[b * 3 + 0], oy = rays_o[b * 3 + 1], oz = rays_o[b * 3 + 2];
    float txx = tx[b * 3 + 0],    tyy = tx[b * 3 + 1],    tzz = tx[b * 3 + 2];

    float T = 1.0f;
    size_t base = ((size_t)b * NR + r) * NS;
    for (int s = 0; s < NS; ++s) {
        float d  = dval(s);
        float ex = txx - (ox + dx * d);
        float ey = tyy - (oy + dy * d);
        float ez = tzz - (oz + dz * d);
        float dist = sqrtf(ex * ex + ey * ey + ez * ez);
        float dl = rintf(dist * (FS_C / SPD_C));               // jnp.round (RNE)
        dl = fminf(fmaxf(dl, 0.0f), (float)(LSEQ - 1));
        DLY[base + s] = (int)dl;

        float ds = (s < NS - 1) ? (dval(s + 1) - dval(s)) : 1e10f;
        float alpha = 1.0f - expf(-attn[base + s] * ds);
        W[base + s] = T * alpha;                               // att_i * alpha
        T *= (1.0f - alpha + 1e-6f);                           // running cumprod
    }
}

// === k2: streaming ray reduction (bandwidth-limited, 302 MB total) =========
__global__ void k2_reduce(const float* __restrict__ sig,
                          const float* __restrict__ W,
                          const int* __restrict__ DLY,
                          float* __restrict__ PART) {
    int rc = blockIdx.x;            // ray chunk 0..15
    int s  = blockIdx.y;            // sample 0..63
    int b  = blockIdx.z;            // batch 0..1
    int t  = threadIdx.x;           // 0..255 -> n = 2t, 2t+1
    int n0 = 2 * t, n1 = 2 * t + 1;

    int r0 = rc * RPC;
    int r1 = r0 + RPC; if (r1 > NR) r1 = NR;

    float a0 = 0.f, a1 = 0.f;
    for (int r = r0; r < r1; ++r) {
        size_t sb = (((size_t)b * NR + r) * NS + s) * (size_t)LSEQ;
        if (r + 1 < r1)   // pull next ray's 2 KB line toward L2 early
            __builtin_prefetch(sig + sb + (size_t)NS * LSEQ + n0, 0, 1);
        size_t wb = ((size_t)b * NR + r) * NS + s;
        float wv = W[wb];           // uniform per block -> scalarized
        int  dly = DLY[wb];
        float2 v = ((const float2*)(sig + sb))[t];   // global_load_b64, coalesced
        a0 += (n0 >= dly) ? wv * v.x : 0.0f;
        a1 += (n1 >= dly) ? wv * v.y : 0.0f;
    }
    size_t o = (((size_t)b * NS + s) * RCHUNK + rc) * (size_t)LSEQ;
    PART[o + n0] = a0;
    PART[o + n1] = a1;
}

// === k3: chunk reduce + tail mask + path loss -> H[128][512] ===============
__global__ void k3_h(const float* __restrict__ PART, float* __restrict__ H) {
    int idx = blockIdx.x * blockDim.x + threadIdx.x;   // 0..65535
    int n = idx & (LSEQ - 1);
    int s = (idx >> 9) & (NS - 1);
    int b = idx >> 15;

    float sum = 0.f;
    size_t base = ((size_t)b * NS + s) * RCHUNK * LSEQ;
    #pragma unroll
    for (int rc = 0; rc < RCHUNK; ++rc) sum += PART[base + (size_t)rc * LSEQ + n];

    int shift = (int)rintf(dval(s) * (FS_C / SPD_C));  // shift_samples[s]
    float h = 0.0f;
    if (n < (LSEQ - 1) - shift) {                      // tail mask
        int i = shift + n;
        int j = (i < 4) ? 5 : i;                       // path_loss[:4]=path_loss[5]
        float pl = 1.0f / ((float)j * (SPD_C / FS_C) + 0.001f);
        h = pl * sum;
    }
    H[((size_t)b * NS + s) * LSEQ + n] = h;
}

// === k3b: precompute DFT trig matrix TRIG[n=512][col=528] ==================
// col in [0,257):   cos(2*pi*col*n/512)
// col in [257,514): sin(2*pi*(col-257)*n/512)     (pad cols -> 0)
// Exact mod-512 angle reduction keeps args tiny (< 2*pi).
__global__ void k3b_trig(float* __restrict__ TRIG) {
    int idx = blockIdx.x * blockDim.x + threadIdx.x;   // 0 .. 512*528-1
    int col = idx % NCOLP;
    int n   = idx / NCOLP;
    if (n >= LSEQ) return;
    const float C = TWOPI / (float)LSEQ;
    float v = 0.0f;
    if (col < LH)        { int kn = (col * n) & (LSEQ - 1);        v = cosf((float)kn * C); }
    else if (col < NCOL) { int kn = ((col - LH) * n) & (LSEQ - 1); v = sinf((float)kn * C); }
    TRIG[(size_t)n * NCOLP + col] = v;
}

// === k4: DFT GEMM on the matrix core: F[128 x 514] = H x TRIG ==============
// One wave32 per 16x16 output tile; grid = (33 n-tiles, 8 m-tiles).
// A (16x4 f32): lanes 0-15 rows M; VGPR pair = {K=0,K=1} (lo) / {K=2,K=3} (hi).
// B (4x16 f32): mirrored striping (N across lanes, K split lo/hi halves).
// Inner loop is pure: 1x b64 A-load + 2x b32 B-loads + 1 WMMA, unrolled x4.
__global__ __launch_bounds__(32) void k4_dft(const float* __restrict__ H,
                                             const float* __restrict__ TRIG,
                                             float* __restrict__ F) {
    int ntile = blockIdx.x;          // 0..32
    int mtile = blockIdx.y;          // 0..7
    int lane  = threadIdx.x;         // 0..31, full wave, EXEC all-ones
    int ll = lane & 15;
    int hi = lane >> 4;
    int row = mtile * 16 + ll;       // m = b*64+s
    int col = ntile * 16 + ll;       // output column j

    const float* hrow = H + (size_t)row * LSEQ;
    const float* tcol = TRIG + col;

    v8f acc = {};
    #pragma unroll 4
    for (int k0 = 0; k0 < LSEQ; k0 += 4) {
        int ka = k0 + 2 * hi;        // this half-wave's K offset
        v2f a;
        a.x = hrow[ka];
        a.y = hrow[ka + 1];
        v2f bb;
        bb.x = tcol[(size_t)ka * NCOLP];
        bb.y = tcol[(size_t)(ka + 1) * NCOLP];
        acc = __builtin_amdgcn_wmma_f32_16x16x4_f32(
            /*neg_a=*/false, a, /*neg_b=*/false, bb,
            /*c_mod=*/(short)0, acc, /*reuse_a=*/false, /*reuse_b=*/false);
    }
    #pragma unroll
    for (int v = 0; v < 8; ++v) {    // D layout: VGPR v -> M = v + 8*hi, N = ll
        int m = mtile * 16 + v + 8 * hi;
        if (col < NCOL) F[(size_t)m * NCOL + col] = acc[v];
    }
}

// === k5: phase rotation + sum over samples -> output (2 x 257 x {re,im}) ===
__global__ void k5_final(const float* __restrict__ F, float* __restrict__ out) {
    int idx = blockIdx.x * blockDim.x + threadIdx.x;
    if (idx >= BS * LH) return;
    int b = idx / LH, k = idx % LH;
    float re = 0.f, im = 0.f;
    for (int s = 0; s < NS; ++s) {
        int m = b * NS + s;
        float Fr =  F[(size_t)m * NCOL + k];           // Σ h cos(2πkn/L)
        float Fi = -F[(size_t)m * NCOL + LH + k];      // -Σ h sin(2πkn/L)
        float p  = dval(s) * (FS_C / SPD_C);           // pts2rx_idx[s]
        float ang = -(TWOPI / (float)LSEQ) * ((float)k * p);
        float cp = cosf(ang), sp = sinf(ang);
        re += Fr * cp - Fi * sp;
        im += Fr * sp + Fi * cp;
    }
    out[(size_t)idx * 2 + 0] = re;
    out[(size_t)idx * 2 + 1] = im;
}

extern "C" void kernel_launch(void* const* d_in, const int* in_sizes, int n_in,
                              void* d_out, int out_size, void* d_ws, size_t ws_size,
                              hipStream_t stream) {
    const float* rays_o = (const float*)d_in[0];
    const float* tx     = (const float*)d_in[1];
    const float* attn   = (const float*)d_in[2];
    const float* sig    = (const float*)d_in[3];

    char* wsb   = (char*)d_ws;
    float* W    = (float*)(wsb + OFF_W);
    int*   DLY  = (int*)  (wsb + OFF_DLY);
    float* PART = (float*)(wsb + OFF_PART);
    float* H    = (float*)(wsb + OFF_H);
    float* F    = (float*)(wsb + OFF_F);
    float* TRIG = (float*)(wsb + OFF_TRIG);
    float* out  = (float*)d_out;

    k1_weights<<<dim3((BS * NR + 127) / 128), 128, 0, stream>>>(rays_o, tx, attn, W, DLY);
    k2_reduce <<<dim3(RCHUNK, NS, BS), 256, 0, stream>>>(sig, W, DLY, PART);
    k3_h      <<<dim3((BS * NS * LSEQ) / 256), 256, 0, stream>>>(PART, H);
    k3b_trig  <<<dim3((LSEQ * NCOLP + 255) / 256), 256, 0, stream>>>(TRIG);
    k4_dft    <<<dim3(33, 8), 32, 0, stream>>>(H, TRIG, F);
    k5_final  <<<dim3((BS * LH + 127) / 128), 128, 0, stream>>>(F, out);
}